// Encoder_15195594293583
// MI455X (gfx1250) — compile-verified
//
#include <hip/hip_runtime.h>
#include <math.h>

// ---------------------------------------------------------------------------
// CDNA5 (gfx1250) implementation of the recurrent GCN encoder.
// All matrix contractions use V_WMMA_F32_16X16X4_F32 (f32-accurate WMMA).
// Sequential scan (2 layers x 71 steps) runs as a chain of small kernels on
// `stream`; all state lives in d_ws / d_out (graph-capture safe).
// ---------------------------------------------------------------------------

typedef float v2f __attribute__((ext_vector_type(2)));
typedef float v8f __attribute__((ext_vector_type(8)));

#define NN    192   // N = W*B rows
#define LSEQ  71    // L = W + O - 1
#define DM    192   // D_MODEL

// WMMA 16x16 f32 tile: C += A(16xK) * B(Kx16), K multiple of 4.
// aRow: per-lane pointer to A[m][0] (m = lane&15), element k at aRow[k*ldaK]
// bCol: per-lane pointer to B[0][n] (n = lane&15), element k at bCol[k*ldbK]
// bScale: 0 to mask invalid B columns (keeps EXEC uniform around WMMA).
static __device__ __forceinline__ v8f wmma_loop(const float* __restrict__ aRow, int ldaK,
                                                const float* __restrict__ bCol, int ldbK,
                                                int K, int koff, float bScale) {
  v8f acc = {};
  for (int k = 0; k < K; k += 4) {
    v2f a, b;
    a.x = aRow[(k + koff) * ldaK];
    a.y = aRow[(k + koff + 1) * ldaK];
    b.x = bCol[(k + koff) * ldbK] * bScale;
    b.y = bCol[(k + koff + 1) * ldbK] * bScale;
    acc = __builtin_amdgcn_wmma_f32_16x16x4_f32(false, a, false, b, (short)0, acc,
                                                false, false);
  }
  return acc;
}

// ------------------------- per-layer precompute ----------------------------

// Adjacency -> Qall (64 x 192): [I | Q1 | Q2] applied on the v-axis.
// a = (softmax(relu(nv1@nv2), axis=1) + I) / 2   (row sums of adj are exactly 2)
// Q1[w][v] = 0.05*d(w,v) + 0.95*a[v][w];  Q2[w][v] = 0.05*d + 0.95*sum_u Q1[w][u]a[v][u]
__global__ void p_adj(const float* __restrict__ nv1, const float* __restrict__ nv2,
                      float* __restrict__ Qall) {
  __shared__ float A[64 * 64];
  __shared__ float Q1[64 * 64];
  int tid = threadIdx.x;
  for (int idx = tid; idx < 4096; idx += 256) {
    int i = idx >> 6, j = idx & 63;
    float s = 0.f;
    for (int k = 0; k < 40; ++k) s += nv1[i * 40 + k] * nv2[k * 64 + j];
    A[idx] = fmaxf(s, 0.f);
  }
  __syncthreads();
  if (tid < 64) {
    float* row = A + tid * 64;
    float m = row[0];
    for (int j = 1; j < 64; ++j) m = fmaxf(m, row[j]);
    float s = 0.f;
    for (int j = 0; j < 64; ++j) { float e = expf(row[j] - m); row[j] = e; s += e; }
    float inv = 0.5f / s;
    for (int j = 0; j < 64; ++j) row[j] *= inv;
    row[tid] += 0.5f;
  }
  __syncthreads();
  for (int idx = tid; idx < 4096; idx += 256) {
    int w = idx >> 6, v = idx & 63;
    float q = 0.95f * A[v * 64 + w] + ((w == v) ? 0.05f : 0.f);
    Q1[w * 64 + v] = q;
    Qall[w * 192 + 64 + v] = q;
    Qall[w * 192 + v] = (w == v) ? 1.f : 0.f;
  }
  __syncthreads();
  for (int idx = tid; idx < 4096; idx += 256) {
    int w = idx >> 6, v = idx & 63;
    float s = 0.f;
    for (int u = 0; u < 64; ++u) s += Q1[w * 64 + u] * A[v * 64 + u];
    Qall[w * 192 + 128 + v] = 0.95f * s + ((w == v) ? 0.05f : 0.f);
  }
}

// Wbig[(i*192+m)][c*64+v] = ws[c][i][m-v] if 0<=m-v<129 else 0   (row-major, ld=1024)
__global__ void p_wbig(const float* __restrict__ ws, float* __restrict__ Wbig, int in_dim) {
  int idx = blockIdx.x * 256 + threadIdx.x;
  int row = idx >> 10, col = idx & 1023;
  int i = row / 192, m = row % 192;
  int c = col >> 6, v = col & 63;
  int k = m - v;
  float val = (k >= 0 && k < 129) ? ws[(c * in_dim + i) * 129 + k] : 0.f;
  Wbig[idx] = val;
}

__global__ void k_zero(float* __restrict__ p, int n) {
  int idx = blockIdx.x * 256 + threadIdx.x;
  if (idx < n) p[idx] = 0.f;
}

// ----------------------------- per-step kernels ----------------------------

// Build xcat[n][Kdim] = [h_row | h_col | x_t]
__global__ void k1_build(float* __restrict__ xcat, const float* __restrict__ hrow,
                         const float* __restrict__ hcol, const float* __restrict__ input,
                         const float* __restrict__ out_all, int t, int layer, int Kdim) {
  int idx = blockIdx.x * 256 + threadIdx.x;
  int n = idx / Kdim, k = idx % Kdim;
  float v;
  if (k < 192) v = hrow[n * 192 + k];
  else if (k < 384) v = hcol[n * 192 + (k - 192)];
  else if (layer == 0) {
    int d = k - 384;
    int w = n >> 3, b = n & 7, o = t - w;
    v = (o >= 0 && o < 48) ? input[((b * 24 + w) * 48 + o) * 192 + d] : 0.f;
  } else {
    v = out_all[(n * LSEQ + t) * 384 + (k - 384)];
  }
  xcat[idx] = v;
}

// H0(192x1024) = xcat(192xKdim) @ Wbig(Kdimx1024) + bs[col>>6]   (tiles 12x64)
__global__ void k2_conv(const float* __restrict__ xcat, const float* __restrict__ Wbig,
                        const float* __restrict__ bs, float* __restrict__ H0, int Kdim) {
  int wave = blockIdx.x * 4 + (threadIdx.x >> 5);
  int lane = threadIdx.x & 31;
  int tm = wave >> 6, tn = wave & 63;
  int l15 = lane & 15;
  int mA = tm * 16 + l15;
  int nB = tn * 16 + l15;
  int koff = (lane >> 4) << 1;
  v8f acc = wmma_loop(xcat + mA * Kdim, 1, Wbig + nB, 1024, Kdim, koff, 1.f);
  int m0 = (lane >> 4) << 3;
  float bias = bs[nB >> 6];
#pragma unroll
  for (int r = 0; r < 8; ++r)
    H0[(tm * 16 + m0 + r) * 1024 + nB] = acc[r] + bias;
}

// HP: (H0 as 3072x64) @ Qall(64x192) -> scatter HP[(n*64+v)*48 + j*16+c]  (tiles 192x12)
__global__ void k3_prop(const float* __restrict__ H0, const float* __restrict__ Qall,
                        float* __restrict__ HP) {
  int wave = blockIdx.x * 4 + (threadIdx.x >> 5);
  int lane = threadIdx.x & 31;
  int tm = wave / 12, tn = wave % 12;
  int l15 = lane & 15;
  int mA = tm * 16 + l15;
  int nB = tn * 16 + l15;
  int koff = (lane >> 4) << 1;
  v8f acc = wmma_loop(H0 + mA * 64, 1, Qall + nB, 192, 64, koff, 1.f);
  int m0 = (lane >> 4) << 3;
  int j = nB >> 6, v = nB & 63;
#pragma unroll
  for (int r = 0; r < 8; ++r) {
    int gr = tm * 16 + m0 + r;
    int n = gr >> 4, c = gr & 15;
    HP[(n * 64 + v) * 48 + j * 16 + c] = acc[r];
  }
}

// HM(12288x32) = gelu(HP(12288x48) @ wm^T + bm)   (tiles 768x2), exact-erf gelu
__global__ void k4_mlp(const float* __restrict__ HP, const float* __restrict__ wm,
                       const float* __restrict__ bm, float* __restrict__ HM) {
  int wave = blockIdx.x * 4 + (threadIdx.x >> 5);
  int lane = threadIdx.x & 31;
  int tm = wave >> 1, tn = wave & 1;
  int l15 = lane & 15;
  int mA = tm * 16 + l15;
  int nB = tn * 16 + l15;
  int koff = (lane >> 4) << 1;
  v8f acc = wmma_loop(HP + mA * 48, 1, wm + nB * 48, 1, 48, koff, 1.f);
  int m0 = (lane >> 4) << 3;
  float bias = bm[nB];
#pragma unroll
  for (int r = 0; r < 8; ++r) {
    float x = acc[r] + bias;
    HM[(tm * 16 + m0 + r) * 32 + nB] = 0.5f * x * (1.f + erff(x * 0.70710678118f));
  }
}

// HE(12288x6) = HM(12288x32) @ we^T + be   (tiles 768x1; cols 6..15 masked)
__global__ void k5_end(const float* __restrict__ HM, const float* __restrict__ we,
                       const float* __restrict__ be, float* __restrict__ HE) {
  int wave = blockIdx.x * 4 + (threadIdx.x >> 5);
  int lane = threadIdx.x & 31;
  int tm = wave;
  int l15 = lane & 15;
  int mA = tm * 16 + l15;
  int bc = (l15 < 6) ? l15 : 5;            // clamp pointer, zero via bScale
  float bScale = (l15 < 6) ? 1.f : 0.f;
  int koff = (lane >> 4) << 1;
  v8f acc = wmma_loop(HM + mA * 32, 1, we + bc * 32, 1, 32, koff, bScale);
  int m0 = (lane >> 4) << 3;
  if (l15 < 6) {
    float bias = be[l15];
#pragma unroll
    for (int r = 0; r < 8; ++r)
      HE[(tm * 16 + m0 + r) * 6 + l15] = acc[r] + bias;
  }
}

// GATE(1152x192) = LayerNorm( (HE rows (n,o)) @ wl^T + bl ) * gamma + beta
// One block = 12 waves = one 16-row tile across all 192 cols; LN fused via LDS.
__global__ void k6_lin_ln(const float* __restrict__ HE, const float* __restrict__ wl,
                          const float* __restrict__ bl, const float* __restrict__ gam,
                          const float* __restrict__ bet, float* __restrict__ GATE) {
  __shared__ float sT[16 * 192];
  __shared__ float sMu[16];
  __shared__ float sRi[16];
  int rt = blockIdx.x;
  int wave = threadIdx.x >> 5, lane = threadIdx.x & 31;
  int l15 = lane & 15;
  int grA = rt * 16 + l15;                  // row = n*6+o
  const float* aRow = HE + (grA / 6) * 384 + (grA % 6);   // k=v at aRow[v*6]
  int gc = wave * 16 + l15;                 // output col d
  const float* bCol = wl + gc * 64;         // wl[d*64+v], k=v at bCol[v]
  int koff = (lane >> 4) << 1;
  v8f acc = wmma_loop(aRow, 6, bCol, 1, 64, koff, 1.f);
  int m0 = (lane >> 4) << 3;
  float bias = bl[gc];
#pragma unroll
  for (int r = 0; r < 8; ++r) sT[(m0 + r) * 192 + gc] = acc[r] + bias;
  __syncthreads();
  if (threadIdx.x < 16) {
    int row = threadIdx.x;
    float s = 0.f;
    for (int d = 0; d < 192; ++d) s += sT[row * 192 + d];
    float mu = s * (1.f / 192.f);
    s = 0.f;
    for (int d = 0; d < 192; ++d) { float df = sT[row * 192 + d] - mu; s += df * df; }
    sMu[row] = mu;
    sRi[row] = rsqrtf(s * (1.f / 192.f) + 1e-5f);
  }
  __syncthreads();
  for (int idx = threadIdx.x; idx < 3072; idx += 384) {
    int row = idx / 192, d = idx % 192;
    GATE[(rt * 16 + row) * 192 + d] = (sT[idx] - sMu[row]) * sRi[row] * gam[d] + bet[d];
  }
}

// Gate nonlinearity + state update + roll + outputs.
__global__ void k7_state(const float* __restrict__ GATE, float* __restrict__ hrow,
                         const float* __restrict__ hcol_cur, float* __restrict__ hcol_nxt,
                         float* __restrict__ out_all, float* __restrict__ out_hr,
                         float* __restrict__ out_hc, int t, int layer) {
  int idx = blockIdx.x * 256 + threadIdx.x;   // 0..36863 = n*192+d
  int n = idx / 192, d = idx % 192;
  const float* g = GATE + (n * 6) * 192 + d;
  float ug_r = 1.f / (1.f + expf(-g[0 * 192]));
  float og_r = 1.f / (1.f + expf(-g[1 * 192]));
  float ug_c = 1.f / (1.f + expf(-g[2 * 192]));
  float og_c = 1.f / (1.f + expf(-g[3 * 192]));
  float ig_r = tanhf(g[4 * 192]);
  float ig_c = tanhf(g[5 * 192]);
  float hr = tanhf((1.f - ug_r) * hrow[idx] + ug_r * ig_r) * og_r;
  float hc = tanhf((1.f - ug_c) * hcol_cur[idx] + ug_c * ig_c) * og_c;
  hrow[idx] = hr;
  hcol_nxt[((n + 8) % 192) * 192 + d] = hc;   // roll(h_col, B=8)
  float* oa = out_all + (n * LSEQ + t) * 384;
  if (layer == 0) { oa[d] = hr; oa[192 + d] = hc; }
  else            { oa[d] += hr; oa[192 + d] += hc; }
  if (t >= 47 && (n >> 3) == (t - 47)) {      // rows: t = O-1+w, group w
    int b = n & 7, w = t - 47;
    out_hr[((b * 2 + layer) * 24 + w) * 192 + d] = hr;
  }
  if (t >= 23 && n >= 184) {                   // cols: t >= W-1, n >= (W-1)*B
    int b = n - 184;
    out_hc[((b * 2 + layer) * 48 + (t - 23)) * 192 + d] = hc;
  }
}

// ------------------------------- host side ---------------------------------

extern "C" void kernel_launch(void* const* d_in, const int* in_sizes, int n_in,
                              void* d_out, int out_size, void* d_ws, size_t ws_size,
                              hipStream_t stream) {
  (void)in_sizes; (void)n_in; (void)out_size; (void)ws_size;
  const float* input = (const float*)d_in[0];

  float* out = (float*)d_out;
  float* out_all = out;                              // (192, 71, 384)
  float* out_hr  = out + 192 * LSEQ * 384;           // (8, 2, 24, 192)
  float* out_hc  = out_hr + 8 * 2 * 24 * 192;        // (8, 2, 48, 192)

  float* wsf  = (float*)d_ws;
  float* Qall = wsf;                 // 64*192      = 12288
  float* Wbig = Qall + 12288;        // 768*1024    = 786432
  float* xcat = Wbig + 786432;       // 192*768     = 147456
  float* H0   = xcat + 147456;       // 192*1024    = 196608
  float* HP   = H0 + 196608;         // 192*64*48   = 589824
  float* HM   = HP + 589824;         // 192*64*32   = 393216
  float* HE   = HM + 393216;         // 192*64*6    = 73728
  float* GATE = HE + 73728;          // 1152*192    = 221184
  float* hrow = GATE + 221184;       // 192*192     = 36864
  float* hcol = hrow + 36864;        // 2 * 36864 (double-buffered, follows hrow)

  for (int li = 0; li < 2; ++li) {
    const float* nv1 = (const float*)d_in[1 + li * 12 + 0];
    const float* nv2 = (const float*)d_in[1 + li * 12 + 1];
    const float* wS  = (const float*)d_in[1 + li * 12 + 2];
    const float* bS  = (const float*)d_in[1 + li * 12 + 3];
    const float* wM  = (const float*)d_in[1 + li * 12 + 4];
    const float* bM  = (const float*)d_in[1 + li * 12 + 5];
    const float* wE  = (const float*)d_in[1 + li * 12 + 6];
    const float* bE  = (const float*)d_in[1 + li * 12 + 7];
    const float* wL  = (const float*)d_in[1 + li * 12 + 8];
    const float* bL  = (const float*)d_in[1 + li * 12 + 9];
    const float* gam = (const float*)d_in[1 + li * 12 + 10];
    const float* bet = (const float*)d_in[1 + li * 12 + 11];

    int in_dim = 3 + li;
    int Kdim = in_dim * 192;

    p_adj<<<1, 256, 0, stream>>>(nv1, nv2, Qall);
    p_wbig<<<(Kdim * 1024) / 256, 256, 0, stream>>>(wS, Wbig, in_dim);
    k_zero<<<(2 * 36864 + 255) / 256, 256, 0, stream>>>(hrow, 2 * 36864); // hrow + hcol buf0

    for (int t = 0; t < LSEQ; ++t) {
      float* hc_cur = hcol + (t & 1) * 36864;
      float* hc_nxt = hcol + ((t + 1) & 1) * 36864;
      k1_build<<<(192 * Kdim) / 256, 256, 0, stream>>>(xcat, hrow, hc_cur, input,
                                                       out_all, t, li, Kdim);
      k2_conv<<<192, 128, 0, stream>>>(xcat, Wbig, bS, H0, Kdim);   // 768 tiles
      k3_prop<<<576, 128, 0, stream>>>(H0, Qall, HP);               // 2304 tiles
      k4_mlp<<<384, 128, 0, stream>>>(HP, wM, bM, HM);              // 1536 tiles
      k5_end<<<192, 128, 0, stream>>>(HM, wE, bE, HE);              // 768 tiles
      k6_lin_ln<<<72, 384, 0, stream>>>(HE, wL, bL, gam, bet, GATE);
      k7_state<<<144, 256, 0, stream>>>(GATE, hrow, hc_cur, hc_nxt, out_all,
                                        out_hr, out_hc, t, li);
    }
  }
}